// GAT_72318659330636
// MI455X (gfx1250) — compile-verified
//
#include <hip/hip_runtime.h>
#include <hip/hip_bf16.h>

typedef __attribute__((ext_vector_type(2))) float v2f;
typedef __attribute__((ext_vector_type(8))) float v8f;

#define NODES   4096
#define FIN     512
#define FOUT    64
#define BATCH   4
#define NEG_BIG (-9000000000000000.0f)
#define ALPHA   0.2f
#define WAVES_PER_BLOCK 8

// ---------------------------------------------------------------------------
// Kernel 1: h = x @ W  (per-wave 16x64 tile via v_wmma_f32_16x16x4_f32),
//           plus fused s1 = h@a1, s2 = h@a2 via butterfly reduction.
// One wave (32 lanes) owns rows [i0, i0+16) of batch b, all 64 output cols.
// ---------------------------------------------------------------------------
__global__ void __launch_bounds__(256) gat_h_kernel(
    const float* __restrict__ x, const float* __restrict__ W,
    const float* __restrict__ a,
    float* __restrict__ h, float* __restrict__ s1, float* __restrict__ s2)
{
    const int lane = threadIdx.x & 31;
    const int wid  = threadIdx.x >> 5;
    const int wg   = blockIdx.x * WAVES_PER_BLOCK + wid;   // 0..1023
    const int b    = wg >> 8;                               // wg / 256
    const int i0   = (wg & 255) << 4;
    const int half = lane >> 4;                             // K-half select
    const int n    = lane & 15;                             // A: row M, B/C: col N

    const float* xrow = x + ((size_t)(b * NODES + i0 + n) * FIN);

    v8f acc0 = {}, acc1 = {}, acc2 = {}, acc3 = {};

    for (int k = 0; k < FIN; k += 4) {
        const int ka = k + 2 * half;
        // A operand: 16x4 f32 tile of x (lane: M = n, K = 2*half + v)
        v2f A; A[0] = xrow[ka]; A[1] = xrow[ka + 1];
        // B operand: 4x16 f32 tiles of W (lane: N = n, K = 2*half + v)
        const float* wr0 = W + (size_t)ka * FOUT + n;
        const float* wr1 = wr0 + FOUT;
        v2f B0; B0[0] = wr0[0];  B0[1] = wr1[0];
        v2f B1; B1[0] = wr0[16]; B1[1] = wr1[16];
        v2f B2; B2[0] = wr0[32]; B2[1] = wr1[32];
        v2f B3; B3[0] = wr0[48]; B3[1] = wr1[48];
        acc0 = __builtin_amdgcn_wmma_f32_16x16x4_f32(false, A, false, B0, (short)0, acc0, false, false);
        acc1 = __builtin_amdgcn_wmma_f32_16x16x4_f32(false, A, false, B1, (short)0, acc1, false, false);
        acc2 = __builtin_amdgcn_wmma_f32_16x16x4_f32(false, A, false, B2, (short)0, acc2, false, false);
        acc3 = __builtin_amdgcn_wmma_f32_16x16x4_f32(false, A, false, B3, (short)0, acc3, false, false);
    }

    // Score vectors for this lane's columns
    const float a10 = a[n],        a11 = a[n + 16], a12 = a[n + 32], a13 = a[n + 48];
    const float a20 = a[FOUT + n], a21 = a[FOUT + n + 16],
                a22 = a[FOUT + n + 32], a23 = a[FOUT + n + 48];

    for (int r = 0; r < 8; ++r) {
        const int row = i0 + r + 8 * half;          // C layout: M = r + 8*(lane/16)
        float* hrow = h + ((size_t)(b * NODES + row) * FOUT) + n;
        const float c0 = acc0[r], c1 = acc1[r], c2 = acc2[r], c3 = acc3[r];
        hrow[0] = c0; hrow[16] = c1; hrow[32] = c2; hrow[48] = c3;

        float p1 = c0 * a10 + c1 * a11 + c2 * a12 + c3 * a13;
        float p2 = c0 * a20 + c1 * a21 + c2 * a22 + c3 * a23;
        // reduce across the 16 lanes that share this row
        for (int m = 1; m < 16; m <<= 1) {
            p1 += __shfl_xor(p1, m, 32);
            p2 += __shfl_xor(p2, m, 32);
        }
        if (n == 0) {
            s1[b * NODES + row] = p1;
            s2[b * NODES + row] = p2;
        }
    }
}

// ---------------------------------------------------------------------------
// Kernel 2: flash-style fused attention.
// Per wave: 16 query rows. Block (8 waves, same batch) stages 64x64 h-tiles
// + s2 slice in LDS. Online softmax with running max/sum; P@h via f32 WMMA.
// ---------------------------------------------------------------------------
__global__ void __launch_bounds__(256) gat_attn_kernel(
    const float* __restrict__ h, const float* __restrict__ s1g,
    const float* __restrict__ s2g, const int* __restrict__ adj,
    float* __restrict__ out)
{
    __shared__ float lds_h[64 * FOUT];
    __shared__ float lds_s2[64];

    const int lane = threadIdx.x & 31;
    const int wid  = threadIdx.x >> 5;
    const int wg   = blockIdx.x * WAVES_PER_BLOCK + wid;   // 0..1023
    const int b    = wg >> 8;
    const int i0   = (wg & 255) << 4;
    const int half = lane >> 4;
    const int n    = lane & 15;                             // A: row M (=m); B/C: col N

    const float sv1 = s1g[b * NODES + i0 + n];              // s1 for this lane's row
    const int* adjrow = adj + (size_t)(i0 + n) * NODES;

    float mrun = NEG_BIG;   // running row max   (valid in lanes: row = n)
    float lrun = 0.0f;      // running row sum
    v8f acc0 = {}, acc1 = {}, acc2 = {}, acc3 = {};

    for (int j0 = 0; j0 < NODES; j0 += 64) {
        __syncthreads();
        // cooperative stage: h[b, j0:j0+64, 0:64] and s2[b, j0:j0+64]
        {
            const float4* src = (const float4*)(h + ((size_t)(b * NODES + j0)) * FOUT);
            float4* dst = (float4*)lds_h;
            for (int t = threadIdx.x; t < 64 * (FOUT / 4); t += 256) dst[t] = src[t];
            if (threadIdx.x < 64) lds_s2[threadIdx.x] = s2g[b * NODES + j0 + threadIdx.x];
        }
        __syncthreads();

        for (int jj = 0; jj < 4; ++jj) {
            const int jb = jj * 16;
            // logits in WMMA A-operand layout: p[2q+c] <-> K = 4q + 2*half + c
            float p[8];
            float mx = NEG_BIG;
            for (int q = 0; q < 4; ++q) {
                for (int c = 0; c < 2; ++c) {
                    const int jl = jb + 4 * q + 2 * half + c;
                    float e = sv1 + lds_s2[jl];
                    e = (e >= 0.0f) ? e : ALPHA * e;          // leaky relu
                    e = (adjrow[j0 + jl] > 0) ? e : NEG_BIG;  // mask
                    p[2 * q + c] = e;
                    mx = fmaxf(mx, e);
                }
            }
            mx = fmaxf(mx, __shfl_xor(mx, 16, 32));           // combine row halves
            const float mnew  = fmaxf(mrun, mx);
            const float scale = __expf(mrun - mnew);
            float rs = 0.0f;
            for (int t = 0; t < 8; ++t) { p[t] = __expf(p[t] - mnew); rs += p[t]; }
            rs += __shfl_xor(rs, 16, 32);
            lrun = lrun * scale + rs;
            mrun = mnew;

            // rescale accumulators: row of acc*[r] is r + 8*half -> stats live in that lane id
            for (int r = 0; r < 8; ++r) {
                const float sr = __shfl(scale, r + 8 * half, 32);
                acc0[r] *= sr; acc1[r] *= sr; acc2[r] *= sr; acc3[r] *= sr;
            }

            // acc += P(16x16) @ h_tile(16x64), as 4 K-steps of 16x16x4 f32 WMMA
            for (int q = 0; q < 4; ++q) {
                v2f A; A[0] = p[2 * q]; A[1] = p[2 * q + 1];
                const float* hr0 = lds_h + (size_t)(jb + 4 * q + 2 * half) * FOUT + n;
                const float* hr1 = hr0 + FOUT;
                v2f B0; B0[0] = hr0[0];  B0[1] = hr1[0];
                v2f B1; B1[0] = hr0[16]; B1[1] = hr1[16];
                v2f B2; B2[0] = hr0[32]; B2[1] = hr1[32];
                v2f B3; B3[0] = hr0[48]; B3[1] = hr1[48];
                acc0 = __builtin_amdgcn_wmma_f32_16x16x4_f32(false, A, false, B0, (short)0, acc0, false, false);
                acc1 = __builtin_amdgcn_wmma_f32_16x16x4_f32(false, A, false, B1, (short)0, acc1, false, false);
                acc2 = __builtin_amdgcn_wmma_f32_16x16x4_f32(false, A, false, B2, (short)0, acc2, false, false);
                acc3 = __builtin_amdgcn_wmma_f32_16x16x4_f32(false, A, false, B3, (short)0, acc3, false, false);
            }
        }
    }

    // finalize: divide by row sum, apply ELU, store
    for (int r = 0; r < 8; ++r) {
        const float lr  = __shfl(lrun, r + 8 * half, 32);
        const float inv = 1.0f / lr;
        const int row   = i0 + r + 8 * half;
        float* o = out + ((size_t)(b * NODES + row)) * FOUT + n;
        float v;
        v = acc0[r] * inv; o[0]  = (v > 0.0f) ? v : expm1f(v);
        v = acc1[r] * inv; o[16] = (v > 0.0f) ? v : expm1f(v);
        v = acc2[r] * inv; o[32] = (v > 0.0f) ? v : expm1f(v);
        v = acc3[r] * inv; o[48] = (v > 0.0f) ? v : expm1f(v);
    }
}

// ---------------------------------------------------------------------------
extern "C" void kernel_launch(void* const* d_in, const int* in_sizes, int n_in,
                              void* d_out, int out_size, void* d_ws, size_t ws_size,
                              hipStream_t stream)
{
    const float* x   = (const float*)d_in[0];   // (4, 4096, 512) f32
    const int*   adj = (const int*)  d_in[1];   // (4096, 4096) i32
    const float* W   = (const float*)d_in[2];   // (512, 64) f32
    const float* a   = (const float*)d_in[3];   // (128, 1) f32
    float* out = (float*)d_out;                 // (4, 4096, 64) f32

    // workspace layout: h (4*4096*64), s1 (4*4096), s2 (4*4096) -- all f32
    float* h  = (float*)d_ws;
    float* s1 = h + (size_t)BATCH * NODES * FOUT;
    float* s2 = s1 + (size_t)BATCH * NODES;

    const int total_waves = BATCH * (NODES / 16);           // 1024
    const int blocks = total_waves / WAVES_PER_BLOCK;       // 128

    gat_h_kernel<<<blocks, 256, 0, stream>>>(x, W, a, h, s1, s2);
    gat_attn_kernel<<<blocks, 256, 0, stream>>>(h, s1, s2, adj, out);
}